// HeteroGNNNetwork_45603962749809
// MI455X (gfx1250) — compile-verified
//
#include <hip/hip_runtime.h>

// ---------------------------------------------------------------------------
// HeteroGNN (2-layer SAGEConv, 6 relations) for MI455X / gfx1250.
//
// Roofline: irregular gather/scatter dominates (~6 GB @ 23.3 TB/s ≈ 0.3 ms);
// dense transforms are ~64 GFLOP total -> done with V_WMMA_F32_16X16X4_F32
// (exact f32, matches reference numerics). Scatter uses native
// global_atomic_add_f32 (relaxed, agent scope).
// ---------------------------------------------------------------------------

typedef __attribute__((ext_vector_type(2))) float v2f;
typedef __attribute__((ext_vector_type(4))) float v4f;
typedef __attribute__((ext_vector_type(8))) float v8f;

static __device__ __forceinline__ void atomAddF32(float* p, float v) {
  __hip_atomic_fetch_add(p, v, __ATOMIC_RELAXED, __HIP_MEMORY_SCOPE_AGENT);
}

// h[n*64+j] = b[j] + sum_k x[n*K+k] * W[j*K+k]      (K = 7 / 4 / 2)
__global__ __launch_bounds__(256) void k_embed(const float* __restrict__ x,
                                               const float* __restrict__ W,
                                               const float* __restrict__ b,
                                               float* __restrict__ h, int K) {
  long t = (long)blockIdx.x * 256 + threadIdx.x;
  int j = (int)(t & 63);
  long n = t >> 6;
  const float* xr = x + n * K;
  const float* wr = W + j * K;
  float acc = b[j];
  for (int k = 0; k < K; ++k) acc = fmaf(xr[k], wr[k], acc);
  h[t] = acc;
}

// 16 lanes per edge; each lane moves 4 consecutive floats (float4 gather,
// 4x global_atomic_add_f32 scatter). Lane 0 bumps the degree counter.
__global__ __launch_bounds__(256) void k_scatter(const int* __restrict__ src,
                                                 const int* __restrict__ dst,
                                                 const float* __restrict__ hsrc,
                                                 float* __restrict__ agg,
                                                 float* __restrict__ cnt) {
  long t = (long)blockIdx.x * 256 + threadIdx.x;
  long e = t >> 4;
  int l = (int)(t & 15);
  int s = src[e];
  int d = dst[e];
  const v4f v = *(const v4f*)(hsrc + (long)s * 64 + l * 4);
  float* p = agg + (long)d * 64 + l * 4;
  atomAddF32(p + 0, v[0]);
  atomAddF32(p + 1, v[1]);
  atomAddF32(p + 2, v[2]);
  atomAddF32(p + 3, v[3]);
  if (l == 0) atomAddF32(cnt + d, 1.0f);
}

// outacc[n][j] += (agg[n]/max(cnt,1)) @ Wl^T + bl + hdst[n] @ Wr^T + br
// One wave per 16-node row block, 4 N-tiles of 16 cols, K=64 in 16 steps of 4.
// Weights staged in LDS with stride 65 (bank-conflict-free B fragments).
__global__ __launch_bounds__(256) void k_sage(const float* __restrict__ agg,
                                              const float* __restrict__ cnt,
                                              const float* __restrict__ hdst,
                                              float* __restrict__ outacc,
                                              const float* __restrict__ Wl,
                                              const float* __restrict__ bl,
                                              const float* __restrict__ Wr,
                                              const float* __restrict__ br) {
  __shared__ float sWl[64 * 65];
  __shared__ float sWr[64 * 65];
  for (int idx = threadIdx.x; idx < 4096; idx += 256) {
    int j = idx >> 6, k = idx & 63;
    sWl[j * 65 + k] = Wl[idx];
    sWr[j * 65 + k] = Wr[idx];
  }
  __syncthreads();

  const int lane = threadIdx.x & 31;
  const int wave = threadIdx.x >> 5;
  const int nl   = lane & 15;     // A: row-in-tile / B: col-in-tile
  const int hi   = lane >> 4;     // 0: K-pair {0,1}, 1: K-pair {2,3}
  const int koff = hi * 2;
  const long mbase = ((long)blockIdx.x * 8 + wave) * 16;
  const long rowA  = mbase + nl;

  const float c  = cnt[rowA];
  const float rc = 1.0f / fmaxf(c, 1.0f);
  const float* aggr = agg  + rowA * 64;
  const float* hr   = hdst + rowA * 64;

  const v8f z = {0.f, 0.f, 0.f, 0.f, 0.f, 0.f, 0.f, 0.f};
  v8f acc[4];
#pragma unroll
  for (int t = 0; t < 4; ++t) acc[t] = z;

  for (int ks = 0; ks < 16; ++ks) {
    const int kk = ks * 4 + koff;
    v2f aA, aH;
    aA[0] = aggr[kk]     * rc;
    aA[1] = aggr[kk + 1] * rc;
    aH[0] = hr[kk];
    aH[1] = hr[kk + 1];
#pragma unroll
    for (int t = 0; t < 4; ++t) {
      const int rbase = (t * 16 + nl) * 65 + kk;   // B[k][n] = W[n][k]
      v2f bL, bR;
      bL[0] = sWl[rbase]; bL[1] = sWl[rbase + 1];
      bR[0] = sWr[rbase]; bR[1] = sWr[rbase + 1];
      acc[t] = __builtin_amdgcn_wmma_f32_16x16x4_f32(
          false, aA, false, bL, (short)0, acc[t], false, false);
      acc[t] = __builtin_amdgcn_wmma_f32_16x16x4_f32(
          false, aH, false, bR, (short)0, acc[t], false, false);
    }
  }

  // C/D layout: vgpr v -> M = v + hi*8, N = nl (per 16-col tile t)
  const long node0 = mbase + hi * 8;
#pragma unroll
  for (int t = 0; t < 4; ++t) {
    const int j = t * 16 + nl;
    const float bias = bl[j] + br[j];
#pragma unroll
    for (int v = 0; v < 8; ++v) {
      float* o = outacc + (node0 + v) * 64 + j;
      *o += acc[t][v] + bias;
    }
  }
}

// out = relu(acc * inv_groupsize)
__global__ __launch_bounds__(256) void k_finalize(const float* __restrict__ acc,
                                                  float* __restrict__ out,
                                                  float inv) {
  long t = (long)blockIdx.x * 256 + threadIdx.x;
  float v = acc[t] * inv;
  out[t] = v > 0.f ? v : 0.f;
}

// q = relu(h @ W1^T + b1) @ W2^T + b2     (64 -> 32 -> 16), one wave per node
__global__ __launch_bounds__(256) void k_head(const float* __restrict__ h,
                                              const float* __restrict__ W1,
                                              const float* __restrict__ b1,
                                              const float* __restrict__ W2,
                                              const float* __restrict__ b2,
                                              float* __restrict__ q) {
  __shared__ float sh[8 * 32];
  const int wave = threadIdx.x >> 5, lane = threadIdx.x & 31;
  const long n = (long)blockIdx.x * 8 + wave;
  const float* hr = h + n * 64;
  const float* w  = W1 + lane * 64;
  float a = b1[lane];
  for (int k = 0; k < 64; ++k) a = fmaf(hr[k], w[k], a);
  sh[wave * 32 + lane] = fmaxf(a, 0.f);
  __syncthreads();
  if (lane < 16) {
    const float* sv = sh + wave * 32;
    const float* w2 = W2 + lane * 32;
    float qv = b2[lane];
    for (int j = 0; j < 32; ++j) qv = fmaf(sv[j], w2[j], qv);
    q[n * 16 + lane] = qv;
  }
}

extern "C" void kernel_launch(void* const* d_in, const int* in_sizes, int n_in,
                              void* d_out, int out_size, void* d_ws, size_t ws_size,
                              hipStream_t stream) {
  (void)in_sizes; (void)n_in; (void)out_size; (void)ws_size;
  const long NA = 262144, NPK = 131072, NL = 524288, E = 1048576;

  const float* x_agv  = (const float*)d_in[0];
  const float* x_pk   = (const float*)d_in[1];
  const float* x_loc  = (const float*)d_in[2];
  const float* We_agv = (const float*)d_in[3];
  const float* be_agv = (const float*)d_in[4];
  const float* We_pk  = (const float*)d_in[5];
  const float* be_pk  = (const float*)d_in[6];
  const float* We_loc = (const float*)d_in[7];
  const float* be_loc = (const float*)d_in[8];
  const float* cWl = (const float*)d_in[9];   // (2,6,64,64)
  const float* cbl = (const float*)d_in[10];  // (2,6,64)
  const float* cWr = (const float*)d_in[11];
  const float* cbr = (const float*)d_in[12];
  const float* aW1 = (const float*)d_in[13];
  const float* ab1 = (const float*)d_in[14];
  const float* aW2 = (const float*)d_in[15];
  const float* ab2 = (const float*)d_in[16];
  const float* pW1 = (const float*)d_in[17];
  const float* pb1 = (const float*)d_in[18];
  const float* pW2 = (const float*)d_in[19];
  const float* pb2 = (const float*)d_in[20];
  const int* srcI[6]; const int* dstI[6];
  for (int r = 0; r < 6; ++r) {
    srcI[r] = (const int*)d_in[21 + 2 * r];
    dstI[r] = (const int*)d_in[22 + 2 * r];
  }

  // workspace layout (floats)
  const long NTOT64 = (NA + NPK + NL) * 64;
  float* bufA = (float*)d_ws;          // layer features, ping
  float* bufB = bufA + NTOT64;         // layer features, pong
  float* agg  = bufB + NTOT64;         // per-relation neighbor sum (max NL rows)
  float* cnt  = agg + NL * 64;         // per-relation degree

  // per-type offsets inside bufA/bufB: 0=agv, 1=picker, 2=loc
  const long tsize[3] = {NA, NPK, NL};
  const long toff[3]  = {0, NA * 64, (NA + NPK) * 64};
  static const int rsrc_t[6] = {0, 2, 0, 1, 0, 1};
  static const int rdst_t[6] = {2, 0, 0, 2, 1, 0};
  const float invg[3] = {1.0f / 3.0f, 1.0f, 0.5f};  // agv:3 rels, picker:1, loc:2

  // output layout: q_agv | q_pk | x_agv | x_pk | x_loc
  float* out    = (float*)d_out;
  float* q_agv  = out;
  float* q_pk   = q_agv + NA * 16;
  float* ox[3];
  ox[0] = q_pk + NPK * 16;       // x_agv
  ox[1] = ox[0] + NA * 64;       // x_picker
  ox[2] = ox[1] + NPK * 64;      // x_loc

  // --- embeddings -> bufA ---
  k_embed<<<(int)(NA * 64 / 256), 256, 0, stream>>>(x_agv, We_agv, be_agv, bufA + toff[0], 7);
  k_embed<<<(int)(NPK * 64 / 256), 256, 0, stream>>>(x_pk, We_pk, be_pk, bufA + toff[1], 4);
  k_embed<<<(int)(NL * 64 / 256), 256, 0, stream>>>(x_loc, We_loc, be_loc, bufA + toff[2], 2);

  // --- 2 HeteroConv layers ---
  for (int l = 0; l < 2; ++l) {
    float* hcur = (l == 0) ? bufA : bufB;
    float* hacc = (l == 0) ? bufB : bufA;
    hipMemsetAsync(hacc, 0, NTOT64 * sizeof(float), stream);

    for (int r = 0; r < 6; ++r) {
      const int st = rsrc_t[r], dt = rdst_t[r];
      const long nd = tsize[dt];
      hipMemsetAsync(agg, 0, nd * 64 * sizeof(float), stream);
      hipMemsetAsync(cnt, 0, nd * sizeof(float), stream);
      k_scatter<<<(int)(E * 16 / 256), 256, 0, stream>>>(
          srcI[r], dstI[r], hcur + toff[st], agg, cnt);
      const long w = ((long)l * 6 + r);
      k_sage<<<(int)(nd / 128), 256, 0, stream>>>(
          agg, cnt, hcur + toff[dt], hacc + toff[dt],
          cWl + w * 64 * 64, cbl + w * 64, cWr + w * 64 * 64, cbr + w * 64);
    }
    for (int t = 0; t < 3; ++t) {
      float* dstp = (l == 0) ? (hacc + toff[t]) : ox[t];  // final layer -> d_out
      k_finalize<<<(int)(tsize[t] * 64 / 256), 256, 0, stream>>>(
          hacc + toff[t], dstp, invg[t]);
    }
  }

  // --- heads (read final features already in d_out) ---
  k_head<<<(int)(NA / 8), 256, 0, stream>>>(ox[0], aW1, ab1, aW2, ab2, q_agv);
  k_head<<<(int)(NPK / 8), 256, 0, stream>>>(ox[1], pW1, pb1, pW2, pb2, q_pk);
}